// BModel_71476845740482
// MI455X (gfx1250) — compile-verified
//
#include <hip/hip_runtime.h>

typedef float v2f __attribute__((ext_vector_type(2)));
typedef float v8f __attribute__((ext_vector_type(8)));

// ---------------------------------------------------------------------------
// helpers
// ---------------------------------------------------------------------------
__device__ __forceinline__ void atomicMaxF(float* addr, float v) {
  // float max via monotone integer mapping (native int atomics, no CAS loop):
  //  v >= 0 : int view monotone increasing  -> atomicMax(int)
  //  v <  0 : uint view monotone decreasing -> atomicMin(uint)
  if (v != v) return;  // NaN: skip
  if (v >= 0.0f) atomicMax((int*)addr, __float_as_int(v));
  else           atomicMin((unsigned int*)addr, __float_as_uint(v));
}

__device__ __forceinline__ int cell64_to_16(int c) {
  int cx = c >> 12, cy = (c >> 6) & 63, cz = c & 63;
  return ((cx >> 2) << 8) | ((cy >> 2) << 4) | (cz >> 2);
}
__device__ __forceinline__ int cell16_to_4(int c) {
  int cx = c >> 8, cy = (c >> 4) & 15, cz = c & 15;
  return ((cx >> 2) << 4) | ((cy >> 2) << 2) | (cz >> 2);
}

__global__ void fill_u32_kernel(unsigned int* __restrict__ p, int n, unsigned int v) {
  int i = blockIdx.x * blockDim.x + threadIdx.x;
  if (i < n) p[i] = v;
}

// ---------------------------------------------------------------------------
// pack kernels: move all the (feature | position | zero-pad) branching out of
// the WMMA inner loop into one trivial elementwise pass.
// Xp[n, k] for k in [0, K_pad):  k < K_feat -> feat; K_feat..K_feat+2 -> pos
// (real coords or grid coords from the node index); else 0.
// ---------------------------------------------------------------------------
__global__ void pack_input_kernel(const float* __restrict__ feat,
                                  const float* __restrict__ pos,
                                  float* __restrict__ Xp,
                                  int N, int K_feat, int K_pad, int grid_bits) {
  long long tid = (long long)blockIdx.x * blockDim.x + threadIdx.x;
  if (tid >= (long long)N * K_pad) return;
  int n = (int)(tid / K_pad);
  int k = (int)(tid % K_pad);
  float v = 0.0f;
  if (k < K_feat) {
    v = feat[(long long)n * K_feat + k];
  } else {
    int kk = k - K_feat;
    if (kk < 3) {
      if (pos) {
        v = pos[n * 3 + kk];
      } else {
        int mask = (1 << grid_bits) - 1;
        int c;
        if (kk == 0)      c = n >> (2 * grid_bits);
        else if (kk == 1) c = (n >> grid_bits) & mask;
        else              c = n & mask;
        v = (float)c;
      }
    }
  }
  Xp[tid] = v;
}

// WpT[o, k] = (k < K_total) ? W[k, o] : 0   (transposed + K-padded)
__global__ void pack_weightT_kernel(const float* __restrict__ W, float* __restrict__ WpT,
                                    int K_total, int K_pad, int OUT) {
  int tid = blockIdx.x * blockDim.x + threadIdx.x;
  if (tid >= OUT * K_pad) return;
  int o = tid / K_pad, k = tid % K_pad;
  WpT[tid] = (k < K_total) ? W[k * OUT + o] : 0.0f;
}

// ---------------------------------------------------------------------------
// z = Xp @ Wp  via V_WMMA_F32_16X16X4_F32. One wave per 16x16 output tile.
// A (16x4 f32): lanes 0-15 row M with K {k,k+1}; lanes 16-31 K {k+2,k+3}.
// B (4x16 f32): mirrored over columns -> contiguous in WpT (transposed).
// C/D: VGPR r -> (M=r, lanes 0-15), (M=r+8, lanes 16-31).
// N and OUT are multiples of 16 -> no guards, EXEC all ones throughout.
// ---------------------------------------------------------------------------
__global__ void __launch_bounds__(128)
node_transform_wmma(const float* __restrict__ Xp, const float* __restrict__ WpT,
                    float* __restrict__ z, int N, int K_pad, int OUT) {
  int lane = threadIdx.x & 31;
  int wave = blockIdx.x * (blockDim.x >> 5) + (threadIdx.x >> 5);
  int mtiles = N >> 4;
  int ntiles = OUT >> 4;
  if (wave >= mtiles * ntiles) return;
  int m0 = (wave % mtiles) << 4;
  int n0 = (wave / mtiles) << 4;

  int half  = lane >> 4;            // 0 | 1
  int l15   = lane & 15;
  int khalf = half << 1;            // K offset 0 | 2
  const float* arow = Xp  + (long long)(m0 + l15) * K_pad + khalf;
  const float* brow = WpT + (long long)(n0 + l15) * K_pad + khalf;

  v8f acc = {0.f, 0.f, 0.f, 0.f, 0.f, 0.f, 0.f, 0.f};
  for (int kb = 0; kb < K_pad; kb += 4) {
    v2f a = *(const v2f*)(arow + kb);   // 8B aligned: K_pad%4==0, khalf even
    v2f b = *(const v2f*)(brow + kb);
    acc = __builtin_amdgcn_wmma_f32_16x16x4_f32(false, a, false, b,
                                                (short)0, acc, false, false);
  }

  int mbase = m0 + (half << 3);
  float* zp = z + (long long)mbase * OUT + n0 + l15;
  for (int r = 0; r < 8; ++r) zp[(long long)r * OUT] = acc[r];
}

// ---------------------------------------------------------------------------
// per-edge segment max: agg[dst] = max(agg[dst], z[src]); one worker per
// (edge, float4 group) -> all lanes of a wave share the same edge indices
// (broadcast loads) and access coalesced row slices.
// level 0: raw ids; 1: 64^3 cell table; 2: table + arithmetic 16^3 remap.
// ---------------------------------------------------------------------------
__global__ void edge_max_kernel(const long long* __restrict__ src,
                                const long long* __restrict__ dst,
                                int E, const int* __restrict__ cell1, int level,
                                const float* __restrict__ z, float* __restrict__ agg,
                                int OUT) {
  int groups = OUT >> 2;
  long long tid = (long long)blockIdx.x * blockDim.x + threadIdx.x;
  if (tid >= (long long)E * groups) return;
  int e = (int)(tid / groups);
  int g = (int)(tid % groups);
  int s = (int)src[e];
  int d = (int)dst[e];
  if (level >= 1) { s = cell1[s]; d = cell1[d]; }
  if (level >= 2) { s = cell64_to_16(s); d = cell64_to_16(d); }
  const float4 zv = *(const float4*)(z + (long long)s * OUT + (g << 2));
  float* ap = agg + (long long)d * OUT + (g << 2);
  atomicMaxF(ap + 0, zv.x);
  atomicMaxF(ap + 1, zv.y);
  atomicMaxF(ap + 2, zv.z);
  atomicMaxF(ap + 3, zv.w);
}

// ---------------------------------------------------------------------------
// out[n,o] = relu( max(agg[n,o], zself[n,o]) - pos[n].W[K_feat:K_feat+3, o] )
// zself==null => pre-pool layer (no self loops); agg==-inf reproduces the
// cnt>0 mask since relu(-inf - p) == 0.  In-place on agg.
// ---------------------------------------------------------------------------
__global__ void finalize_conv(float* __restrict__ agg, const float* __restrict__ zself,
                              const float* __restrict__ W, int K_feat, int OUT,
                              const float* __restrict__ pos, int grid_bits, int N) {
  long long tid = (long long)blockIdx.x * blockDim.x + threadIdx.x;
  if (tid >= (long long)N * OUT) return;
  int n = (int)(tid / OUT);
  int o = (int)(tid % OUT);
  float px, py, pz;
  if (pos) {
    px = pos[n * 3 + 0]; py = pos[n * 3 + 1]; pz = pos[n * 3 + 2];
  } else {
    int mask = (1 << grid_bits) - 1;
    px = (float)(n >> (2 * grid_bits));
    py = (float)((n >> grid_bits) & mask);
    pz = (float)(n & mask);
  }
  float p = px * W[(K_feat + 0) * OUT + o]
          + py * W[(K_feat + 1) * OUT + o]
          + pz * W[(K_feat + 2) * OUT + o];
  float m = agg[tid];
  if (zself) m = fmaxf(m, zself[tid]);
  agg[tid] = fmaxf(0.0f, m - p);
}

// ---------------------------------------------------------------------------
// pool 1: nodes (float coords) -> 64^3 cells; writes cell table + cell max.
// f1 pre-zeroed (inputs are post-ReLU >= 0, reproducing the cnt>0 mask).
// ---------------------------------------------------------------------------
__global__ void pool1_kernel(const float* __restrict__ nodes, const float* __restrict__ x,
                             int* __restrict__ cell1, float* __restrict__ f1,
                             int N, int OUT) {
  int tid = blockIdx.x * blockDim.x + threadIdx.x;
  if (tid >= N * OUT) return;
  int n = tid / OUT, o = tid % OUT;
  int cx = (int)floorf(nodes[n * 3 + 0] * 0.25f);
  int cy = (int)floorf(nodes[n * 3 + 1] * 0.25f);
  int cz = (int)floorf(nodes[n * 3 + 2] * 0.25f);
  cx = cx < 0 ? 0 : (cx > 63 ? 63 : cx);
  cy = cy < 0 ? 0 : (cy > 63 ? 63 : cy);
  cz = cz < 0 ? 0 : (cz > 63 ? 63 : cz);
  int c = (cx << 12) | (cy << 6) | cz;
  if (o == 0) cell1[n] = c;
  atomicMaxF(&f1[(long long)c * OUT + o], x[tid]);
}

// grid->coarser-grid max pooling (which==2: 64^3->16^3, which==3: 16^3->4^3)
__global__ void pool_grid_kernel(const float* __restrict__ x, float* __restrict__ f,
                                 int N, int OUT, int which) {
  long long tid = (long long)blockIdx.x * blockDim.x + threadIdx.x;
  if (tid >= (long long)N * OUT) return;
  int c = (int)(tid / OUT), o = (int)(tid % OUT);
  int cc = (which == 2) ? cell64_to_16(c) : cell16_to_4(c);
  atomicMaxF(&f[(long long)cc * OUT + o], x[tid]);
}

// out[j] = sum_k pooled[k] * Wlin[k, j]   (K=8192, M=100)
__global__ void final_linear_kernel(const float* __restrict__ pooled,
                                    const float* __restrict__ Wlin,
                                    float* __restrict__ out, int K, int M) {
  __shared__ float red[256];
  int j = blockIdx.x;
  float s = 0.0f;
  for (int k = threadIdx.x; k < K; k += blockDim.x)
    s += pooled[k] * Wlin[(long long)k * M + j];
  red[threadIdx.x] = s;
  __syncthreads();
  for (int w = blockDim.x >> 1; w > 0; w >>= 1) {
    if ((int)threadIdx.x < w) red[threadIdx.x] += red[threadIdx.x + w];
    __syncthreads();
  }
  if (threadIdx.x == 0) out[j] = red[0];
}

// ---------------------------------------------------------------------------
extern "C" void kernel_launch(void* const* d_in, const int* in_sizes, int n_in,
                              void* d_out, int out_size, void* d_ws, size_t ws_size,
                              hipStream_t stream) {
  (void)n_in; (void)ws_size;
  const float*     nodes    = (const float*)d_in[0];
  const float*     features = (const float*)d_in[1];
  const long long* edges    = (const long long*)d_in[2];
  const float* W1 = (const float*)d_in[3];
  const float* W2 = (const float*)d_in[4];
  const float* W3 = (const float*)d_in[5];
  const float* W4 = (const float*)d_in[6];
  const float* W5 = (const float*)d_in[7];
  const float* W6 = (const float*)d_in[8];
  const float* W7 = (const float*)d_in[9];
  const float* Wlin = (const float*)d_in[10];

  const int E = in_sizes[2] / 2;
  const long long* esrc = edges;
  const long long* edst = edges + E;

  // ---- workspace: big slabs reused by time-disjoint tenants ----
  char* ws = (char*)d_ws;
  size_t off = 0;
  auto alloc = [&](size_t bytes) -> void* {
    void* p = ws + off;
    off = (off + bytes + 255) & ~(size_t)255;
    return p;
  };
  float* Xp    = (float*)alloc(262144ull * 36 * 4);   // packed inputs (max layer)
  float* slabB = (float*)alloc(262144ull * 32 * 4);   // z1/x1/z2/x2, later z3/z4
  float* slabC = (float*)alloc(262144ull * 32 * 4);   // x3, later post-pool smalls
  float* slabD = (float*)alloc(262144ull * 32 * 4);   // f1, later x4
  int*   cell1 = (int*)  alloc(65536ull * 4);
  float* WpT   = (float*)alloc(128ull * 68 * 4);      // packed transposed weights

  // slabB tenants (pre-pool phase; dead before z3 is written)
  float* z1 = slabB;                   // 65536*8
  float* x1 = slabB +  524288;         // 65536*8
  float* z2 = slabB + 1048576;         // 65536*16
  float* x2 = slabB + 2097152;         // 65536*16
  float* z3 = slabB;                   // 262144*32 (after pool1)
  float* z4 = slabB;                   // reuse (z3 dead after L3 finalize)
  float* x3 = slabC;                   // 262144*32
  float* f1 = slabD;                   // 262144*16 (dead after L3 pack)
  float* x4 = slabD;                   // 262144*32 (after f1 dead)
  // slabC tenants after x3 dead (post L4 pack)
  float* f2     = slabC;               // 4096*32
  float* z5     = slabC + 131072;      // 4096*64
  float* x5     = z5 + 262144;
  float* z6     = x5 + 262144;
  float* x6     = z6 + 262144;
  float* z7     = x6 + 262144;         // 4096*128
  float* x7     = z7 + 524288;
  float* pooled = x7 + 524288;         // 64*128

  const unsigned NEG_INF = 0xFF800000u;
  auto fill = [&](void* p, long long n, unsigned v) {
    int blocks = (int)((n + 255) / 256);
    fill_u32_kernel<<<blocks, 256, 0, stream>>>((unsigned*)p, (int)n, v);
  };
  // pack inputs + weights, then dense WMMA transform: z = [feat,pos] @ W
  auto transform = [&](const float* feat, const float* pos, const float* W, float* z,
                       int N, int K_feat, int K_total, int OUT, int gbits) {
    int K_pad = (K_total + 3) & ~3;
    long long pw = (long long)N * K_pad;
    pack_input_kernel<<<(int)((pw + 255) / 256), 256, 0, stream>>>(
        feat, pos, Xp, N, K_feat, K_pad, gbits);
    int ww = OUT * K_pad;
    pack_weightT_kernel<<<(ww + 255) / 256, 256, 0, stream>>>(W, WpT, K_total, K_pad, OUT);
    int waves = (N / 16) * (OUT / 16);
    node_transform_wmma<<<(waves + 3) / 4, 128, 0, stream>>>(Xp, WpT, z, N, K_pad, OUT);
  };
  auto edge = [&](int level, const float* z, float* agg, int OUT) {
    long long work = (long long)E * (OUT / 4);
    edge_max_kernel<<<(int)((work + 255) / 256), 256, 0, stream>>>(
        esrc, edst, E, cell1, level, z, agg, OUT);
  };
  auto finalize = [&](float* agg, const float* zself, const float* W, int K_feat,
                      int OUT, const float* pos, int gbits, int N) {
    long long work = (long long)N * OUT;
    finalize_conv<<<(int)((work + 255) / 256), 256, 0, stream>>>(
        agg, zself, W, K_feat, OUT, pos, gbits, N);
  };

  // ---- layer 1 (65536 nodes, real coords, 1->8) ----
  transform(features, nodes, W1, z1, 65536, 1, 4, 8, 0);
  fill(x1, 65536ll * 8, NEG_INF);
  edge(0, z1, x1, 8);
  finalize(x1, nullptr, W1, 1, 8, nodes, 0, 65536);
  // ---- layer 2 (8->16) ----
  transform(x1, nodes, W2, z2, 65536, 8, 11, 16, 0);
  fill(x2, 65536ll * 16, NEG_INF);
  edge(0, z2, x2, 16);
  finalize(x2, nullptr, W2, 8, 16, nodes, 0, 65536);
  // ---- pool 1: 65536 nodes -> 64^3 cells ----
  fill(f1, 262144ll * 16, 0u);
  pool1_kernel<<<(65536 * 16 + 255) / 256, 256, 0, stream>>>(nodes, x2, cell1, f1, 65536, 16);
  // ---- layer 3 (64^3, 16->32) ----
  transform(f1, nullptr, W3, z3, 262144, 16, 19, 32, 6);
  fill(x3, 262144ll * 32, NEG_INF);
  edge(1, z3, x3, 32);
  finalize(x3, z3, W3, 16, 32, nullptr, 6, 262144);
  // ---- layer 4 (32->32) ----
  transform(x3, nullptr, W4, z4, 262144, 32, 35, 32, 6);
  fill(x4, 262144ll * 32, NEG_INF);
  edge(1, z4, x4, 32);
  finalize(x4, z4, W4, 32, 32, nullptr, 6, 262144);
  // ---- pool 2: 64^3 -> 16^3 ----
  fill(f2, 4096ll * 32, 0u);
  pool_grid_kernel<<<(int)((262144ll * 32 + 255) / 256), 256, 0, stream>>>(x4, f2, 262144, 32, 2);
  // ---- layer 5 (16^3, 32->64) ----
  transform(f2, nullptr, W5, z5, 4096, 32, 35, 64, 4);
  fill(x5, 4096ll * 64, NEG_INF);
  edge(2, z5, x5, 64);
  finalize(x5, z5, W5, 32, 64, nullptr, 4, 4096);
  // ---- layer 6 (64->64) ----
  transform(x5, nullptr, W6, z6, 4096, 64, 67, 64, 4);
  fill(x6, 4096ll * 64, NEG_INF);
  edge(2, z6, x6, 64);
  finalize(x6, z6, W6, 64, 64, nullptr, 4, 4096);
  // ---- layer 7 (64->128) ----
  transform(x6, nullptr, W7, z7, 4096, 64, 67, 128, 4);
  fill(x7, 4096ll * 128, NEG_INF);
  edge(2, z7, x7, 128);
  finalize(x7, z7, W7, 64, 128, nullptr, 4, 4096);
  // ---- output pool 16^3 -> 4^3, then linear 8192 -> 100 ----
  fill(pooled, 64ll * 128, 0u);
  pool_grid_kernel<<<(int)((4096ll * 128 + 255) / 256), 256, 0, stream>>>(x7, pooled, 4096, 128, 3);
  final_linear_kernel<<<out_size, 256, 0, stream>>>(pooled, Wlin, (float*)d_out, 8192, out_size);
}